// PredictionAwareSAE_80393197846567
// MI455X (gfx1250) — compile-verified
//
#include <hip/hip_runtime.h>
#include <hip/hip_bf16.h>
#include <cstdint>

typedef float v2f __attribute__((ext_vector_type(2)));
typedef float v8f __attribute__((ext_vector_type(8)));

// ---------------------------------------------------------------------------
// Kernel 1: pre_act[B,H] = (x - pre_bias) @ W^T + latent_bias   (fp32 WMMA)
// Block tile 64(M) x 128(N), K-chunk 32. 256 threads = 8 waves (wave32),
// wave grid 2x4, each wave owns a 32x32 patch = 2x2 WMMA 16x16 tiles.
// ---------------------------------------------------------------------------
__global__ __launch_bounds__(256) void sae_enc_gemm(
    const float* __restrict__ x, const float* __restrict__ W,
    const float* __restrict__ pre_bias, const float* __restrict__ latent_bias,
    float* __restrict__ preact, int B, int D, int H)
{
    constexpr int TM = 64, TN = 128, TK = 32, LD = TK + 4; // pad->stride 36 floats
    __shared__ float Xs[TM * LD];
    __shared__ float Ws[TN * LD];

    const int tid   = threadIdx.x;
    const int lane  = tid & 31;
    const int wv    = tid >> 5;
    const int waveM = wv >> 2;     // 0..1
    const int waveN = wv & 3;      // 0..3
    const int bm    = blockIdx.y * TM;
    const int bn    = blockIdx.x * TN;
    const int l15   = lane & 15;
    const int khi   = (lane >> 4) << 1;   // 0 or 2: K-half select per A/B layout

    v8f acc[2][2] = {};

    for (int k0 = 0; k0 < D; k0 += TK) {
        // ---- stage X tile (64x32) minus pre_bias, float4 traffic ----
        #pragma unroll
        for (int i = tid; i < (TM * TK) / 4; i += 256) {
            const int r = i >> 3;            // 8 float4 per row
            const int c = (i & 7) << 2;
            const float4 xv = *reinterpret_cast<const float4*>(
                &x[(size_t)(bm + r) * D + k0 + c]);
            const float4 pb = *reinterpret_cast<const float4*>(&pre_bias[k0 + c]);
            float4 v; v.x = xv.x - pb.x; v.y = xv.y - pb.y;
                      v.z = xv.z - pb.z; v.w = xv.w - pb.w;
            *reinterpret_cast<float4*>(&Xs[r * LD + c]) = v;
        }
        // ---- stage W tile (128x32) ----
        #pragma unroll
        for (int i = tid; i < (TN * TK) / 4; i += 256) {
            const int r = i >> 3;
            const int c = (i & 7) << 2;
            *reinterpret_cast<float4*>(&Ws[r * LD + c]) =
                *reinterpret_cast<const float4*>(&W[(size_t)(bn + r) * D + k0 + c]);
        }
        __syncthreads();

        #pragma unroll
        for (int kk = 0; kk < TK; kk += 4) {
            const int kh = kk + khi;
            v2f a[2], b[2];
            #pragma unroll
            for (int mi = 0; mi < 2; ++mi) {
                const int r = waveM * 32 + mi * 16 + l15;
                a[mi] = *reinterpret_cast<const v2f*>(&Xs[r * LD + kh]);
            }
            #pragma unroll
            for (int ni = 0; ni < 2; ++ni) {
                const int r = waveN * 32 + ni * 16 + l15;
                b[ni] = *reinterpret_cast<const v2f*>(&Ws[r * LD + kh]);
            }
            #pragma unroll
            for (int mi = 0; mi < 2; ++mi)
                #pragma unroll
                for (int ni = 0; ni < 2; ++ni)
                    acc[mi][ni] = __builtin_amdgcn_wmma_f32_16x16x4_f32(
                        false, a[mi], false, b[ni],
                        (short)0, acc[mi][ni], false, false);
        }
        __syncthreads();
    }

    // ---- epilogue: C layout: VGPR r -> M = r + 8*(lane>>4); N = lane&15 ----
    #pragma unroll
    for (int mi = 0; mi < 2; ++mi) {
        #pragma unroll
        for (int ni = 0; ni < 2; ++ni) {
            const int col = bn + waveN * 32 + ni * 16 + l15;
            const float lb = latent_bias[col];
            #pragma unroll
            for (int r = 0; r < 8; ++r) {
                const int row = bm + waveM * 32 + mi * 16 + r + ((lane >> 4) << 3);
                preact[(size_t)row * H + col] = acc[mi][ni][r] + lb;
            }
        }
    }
}

// ---------------------------------------------------------------------------
// Kernel 2: per-row top-k via bisection on value threshold. Row cached in
// dynamic LDS (H*4 = 96KB; WGP has 320KB). Writes dense relu'd features
// in place and a compacted (idx,val) list to workspace.
// ---------------------------------------------------------------------------
__global__ __launch_bounds__(256) void sae_topk(
    float* __restrict__ feat, const int* __restrict__ kptr,
    int* __restrict__ cnt, int* __restrict__ idxs, float* __restrict__ vals,
    int H)
{
    extern __shared__ float smem[];
    float* rowbuf = smem;          // H floats
    float* red    = smem + H;      // 256 floats (also reused as int)
    int*   redi   = reinterpret_cast<int*>(red);
    __shared__ int   nsel;
    __shared__ float s_lo, s_hi;

    const int row = blockIdx.x;
    const int tid = threadIdx.x;
    const float* src = feat + (size_t)row * H;

    // load row (float4 coalesced)
    const float4* src4 = reinterpret_cast<const float4*>(src);
    float4* buf4 = reinterpret_cast<float4*>(rowbuf);
    for (int i = tid; i < H / 4; i += 256) buf4[i] = src4[i];
    __syncthreads();

    // row min / max
    float lmax = -3.4e38f, lmin = 3.4e38f;
    for (int i = tid; i < H; i += 256) {
        const float v = rowbuf[i];
        lmax = fmaxf(lmax, v); lmin = fminf(lmin, v);
    }
    red[tid] = lmax; __syncthreads();
    for (int s = 128; s > 0; s >>= 1) {
        if (tid < s) red[tid] = fmaxf(red[tid], red[tid + s]);
        __syncthreads();
    }
    if (tid == 0) s_hi = red[0];
    __syncthreads();
    red[tid] = lmin; __syncthreads();
    for (int s = 128; s > 0; s >>= 1) {
        if (tid < s) red[tid] = fminf(red[tid], red[tid + s]);
        __syncthreads();
    }
    if (tid == 0) s_lo = red[0];
    __syncthreads();

    const int K = kptr[0];
    float lo = s_lo, hi = s_hi;    // invariant: count(>=lo) >= K
    for (int it = 0; it < 32; ++it) {
        const float mid = 0.5f * (lo + hi);
        int c = 0;
        for (int i = tid; i < H; i += 256) c += (rowbuf[i] >= mid) ? 1 : 0;
        redi[tid] = c; __syncthreads();
        for (int s = 128; s > 0; s >>= 1) {
            if (tid < s) redi[tid] += redi[tid + s];
            __syncthreads();
        }
        const int total = redi[0];
        __syncthreads();
        if (total >= K) lo = mid; else hi = mid;
    }

    if (tid == 0) nsel = 0;
    __syncthreads();
    const float th = lo;
    for (int i = tid; i < H; i += 256) {
        const float v = rowbuf[i];
        const bool sel = (v >= th);
        feat[(size_t)row * H + i] = sel ? fmaxf(v, 0.f) : 0.f;
        if (sel) {
            const int slot = atomicAdd(&nsel, 1);
            if (slot < 64) {
                idxs[row * 64 + slot] = i;
                vals[row * 64 + slot] = fmaxf(v, 0.f);
            }
        }
    }
    __syncthreads();
    if (tid == 0) cnt[row] = nsel < 64 ? nsel : 64;
}

// ---------------------------------------------------------------------------
// Kernel 3: sparse decode  x_hat[b,:] = sum_j val_j * W[idx_j,:] + pre_bias
// One block per row; W rows are L2-resident (75MB < 192MB L2).
// ---------------------------------------------------------------------------
__global__ __launch_bounds__(256) void sae_dec(
    const float* __restrict__ W, const float* __restrict__ pre_bias,
    const int* __restrict__ cnt, const int* __restrict__ idxs,
    const float* __restrict__ vals, float* __restrict__ xhat, int D)
{
    const int row = blockIdx.x, tid = threadIdx.x;
    __shared__ int   s_idx[64];
    __shared__ float s_val[64];
    __shared__ int   s_n;
    if (tid == 0) s_n = cnt[row];
    if (tid < 64) { s_idx[tid] = idxs[row * 64 + tid]; s_val[tid] = vals[row * 64 + tid]; }
    __syncthreads();
    const int n = s_n;
    float a0 = 0.f, a1 = 0.f, a2 = 0.f;
    for (int j = 0; j < n; ++j) {
        const float* wr = W + (size_t)s_idx[j] * D;
        const float v = s_val[j];
        a0 = fmaf(v, wr[tid], a0);
        a1 = fmaf(v, wr[tid + 256], a1);
        a2 = fmaf(v, wr[tid + 512], a2);
    }
    xhat[(size_t)row * D + tid]       = a0 + pre_bias[tid];
    xhat[(size_t)row * D + tid + 256] = a1 + pre_bias[tid + 256];
    xhat[(size_t)row * D + tid + 512] = a2 + pre_bias[tid + 512];
}

// ---------------------------------------------------------------------------
extern "C" void kernel_launch(void* const* d_in, const int* in_sizes, int n_in,
                              void* d_out, int out_size, void* d_ws, size_t ws_size,
                              hipStream_t stream) {
    const float* x           = (const float*)d_in[0];
    const float* W           = (const float*)d_in[1];
    const float* pre_bias    = (const float*)d_in[2];
    const float* latent_bias = (const float*)d_in[3];
    const int*   kptr        = (const int*)d_in[4];

    const int D = in_sizes[2];           // 768
    const int H = in_sizes[3];           // 24576
    const int B = in_sizes[0] / D;       // 8192

    float* xhat = (float*)d_out;
    float* feat = xhat + (size_t)B * D;  // features region (staged pre_act first)

    int*   cnt  = (int*)d_ws;
    int*   idxs = cnt + B;
    float* vals = (float*)(idxs + (size_t)B * 64);

    // 1) encoder GEMM -> pre_act staged in feat region
    dim3 g1(H / 128, B / 64);
    sae_enc_gemm<<<g1, 256, 0, stream>>>(x, W, pre_bias, latent_bias, feat, B, D, H);

    // 2) per-row top-k (row cached in dynamic LDS)
    const size_t shmem = (size_t)H * sizeof(float) + 256 * sizeof(float);
    sae_topk<<<B, 256, shmem, stream>>>(feat, kptr, cnt, idxs, vals, H);

    // 3) sparse decode
    sae_dec<<<B, 256, 0, stream>>>(W, pre_bias, cnt, idxs, vals, xhat, D);
}